// LearnPose_64647847740146
// MI455X (gfx1250) — compile-verified
//
#include <hip/hip_runtime.h>

typedef __attribute__((ext_vector_type(2))) float v2f;
typedef __attribute__((ext_vector_type(8))) float v8f;

// out[g] = init_c2w[idx[g]] @ [[exp(r[idx[g]]), t[idx[g]]],[0 0 0 1]]
// 4 matrices per wave per iteration via V_WMMA_F32_16X16X4_F32 block packing.
// Epilogue: D diagonal-block columns -> LDS tile -> one coalesced b128 store.
__global__ __launch_bounds__(256) void pose_wmma_kernel(
    const float* __restrict__ r,        // (N,3)
    const float* __restrict__ t,        // (N,3)
    const float* __restrict__ init_c2w, // (N,4,4)
    const int*   __restrict__ frame_idx,// (M,) int32 (JAX x64 off)
    float*       __restrict__ out,      // (M,4,4)
    int M, int nGroups)
{
    // 8 waves/block * 4 matrices * 16 floats = 2 KB staging tile
    __shared__ __align__(16) float lds[8 * 64];

    const int lane          = threadIdx.x & 31;
    const int wavesPerBlock = blockDim.x >> 5;
    const int wave          = blockIdx.x * wavesPerBlock + (threadIdx.x >> 5);
    const int numWaves      = gridDim.x * wavesPerBlock;
    float*    ldsw          = lds + (threadIdx.x >> 5) * 64;

    const int hi  = lane >> 4;   // 0: K rows 0-1, 1: K rows 2-3
    const int l   = lane & 15;
    const int blk = l >> 2;      // which of the 4 matrices in this group
    const int q   = l & 3;       // A: row within 4x4 block; B: column within block

    // D diagonal-block writer mapping:
    //   lanes 0-7   -> blocks 0,1 (acc VGPRs 0-3 / 4-7)
    //   lanes 24-31 -> blocks 2,3 (acc VGPRs 0-3 / 4-7)
    const bool writer = (lane < 8) || (lane >= 24);
    const int  wblk   = (lane < 8) ? (lane >> 2) : (2 + ((lane - 24) >> 2));
    const int  wcol   = lane & 3;
    const bool whalf  = (wblk & 1);  // use D[4..7] instead of D[0..3]

    for (int g = wave; g < nGroups; g += numWaves) {
        const int base = g << 2;
        const int gi   = base + blk;
        const int gic  = (gi < M) ? gi : (M - 1);       // clamp; store masked later
        const int idxv = frame_idx[gic];

        // ---- A operand: row q of init_c2w[idxv], half 'hi' (float2, coalesced) ----
        const float* ip = init_c2w + (((long long)idxv) << 4) + (q << 2) + (hi << 1);
        v2f A;
        A.x = ip[0];
        A.y = ip[1];

        // ---- Rodrigues inputs (r[0]/t[0] forced to zero, branchless) ----
        const float  zmask = (idxv != 0) ? 1.0f : 0.0f;
        const float* rp = r + (long long)idxv * 3;
        const float* tp = t + (long long)idxv * 3;
        const float x  = rp[0] * zmask, y  = rp[1] * zmask, z  = rp[2] * zmask;
        const float tx = tp[0] * zmask, ty = tp[1] * zmask, tz = tp[2] * zmask;

        const float n2  = x * x + y * y + z * z;
        const float nrm = __builtin_sqrtf(n2) + 1e-15f;
        const float sn  = __sinf(nrm) / nrm;                   // sin(n)/n
        const float cs  = (1.0f - __cosf(nrm)) / (nrm * nrm);  // (1-cos(n))/n^2

        // R = I + sn*K + cs*K^2
        const float xy = x * y, xz = x * z, yz = y * z;
        const float R00 = 1.0f - cs * (y * y + z * z);
        const float R01 = -sn * z + cs * xy;
        const float R02 =  sn * y + cs * xz;
        const float R10 =  sn * z + cs * xy;
        const float R11 = 1.0f - cs * (x * x + z * z);
        const float R12 = -sn * x + cs * yz;
        const float R20 = -sn * y + cs * xz;
        const float R21 =  sn * x + cs * yz;
        const float R22 = 1.0f - cs * (x * x + y * y);

        // ---- B operand: c2w[k][q] for k = 2*hi, 2*hi+1 (branchless selects) ----
        const float row0 = (q == 0) ? R00 : (q == 1) ? R01 : (q == 2) ? R02 : tx;
        const float row1 = (q == 0) ? R10 : (q == 1) ? R11 : (q == 2) ? R12 : ty;
        const float row2 = (q == 0) ? R20 : (q == 1) ? R21 : (q == 2) ? R22 : tz;
        const float row3 = (q == 3) ? 1.0f : 0.0f;
        v2f B;
        B.x = hi ? row2 : row0;
        B.y = hi ? row3 : row1;

        // ---- 4 independent 4x4 matmuls in one WMMA (EXEC all-1s here) ----
        v8f C = {};
        v8f D = __builtin_amdgcn_wmma_f32_16x16x4_f32(
            /*neg_a=*/false, A, /*neg_b=*/false, B,
            /*c_mod=*/(short)0, C, /*reuse_a=*/false, /*reuse_b=*/false);

        // ---- stage diagonal blocks (per-lane columns) into the wave's LDS tile ----
        // banks 16*wblk + 4*i + wcol are all distinct per store: conflict-free
        if (writer) {
            float* wp = ldsw + (wblk << 4) + wcol;
            wp[0]  = whalf ? D[4] : D[0];
            wp[4]  = whalf ? D[5] : D[1];
            wp[8]  = whalf ? D[6] : D[2];
            wp[12] = whalf ? D[7] : D[3];
        }

        // ---- same-wave readback (DS ops are in-order; compiler waits DScnt):
        //      lane n (n<16) holds row n%4 of matrix n/4 -> one coalesced b128
        //      store covering the group's contiguous 256 B of output.
        if (lane < 16) {
            const int og = base + (lane >> 2);
            if (og < M) {
                const float4 row = *(const float4*)(ldsw + (lane << 2));
                *(float4*)(out + ((long long)base << 4) + (lane << 2)) = row;
            }
        }
    }
}

extern "C" void kernel_launch(void* const* d_in, const int* in_sizes, int n_in,
                              void* d_out, int out_size, void* d_ws, size_t ws_size,
                              hipStream_t stream) {
    const float* r         = (const float*)d_in[0];
    const float* t         = (const float*)d_in[1];
    const float* init_c2w  = (const float*)d_in[2];
    const int*   frame_idx = (const int*)d_in[3];
    float*       out       = (float*)d_out;

    const int M       = in_sizes[3];
    const int nGroups = (M + 3) >> 2;

    const int threads       = 256;               // 8 wave32 waves per block
    const int wavesPerBlock = threads / 32;
    long long blocks = ((long long)nGroups + wavesPerBlock - 1) / wavesPerBlock;
    if (blocks > 262144) blocks = 262144;        // grid-stride covers the rest
    if (blocks < 1) blocks = 1;

    pose_wmma_kernel<<<(int)blocks, threads, 0, stream>>>(
        r, t, init_c2w, frame_idx, out, M, nGroups);
}